// S4Layer_3891240370281
// MI455X (gfx1250) — compile-verified
//
#include <hip/hip_runtime.h>
#include <math.h>

// ---------------------------------------------------------------------------
// S4 layer forward for MI455X (gfx1250).
//   Phase A: Cauchy-kernel GEMM via V_WMMA_F32_16X16X4_F32 (fp32 WMMA).
//   Phase B: IFFT(at_roots) -> time kernel K      (LDS Stockham FFT-4096)
//   Phase C: Kf = rfft(K, 8192)                   (real-packed FFT-4096)
//   Phase D: y = irfft(rfft(x,8192) * Kf)[:4096]  (real-packed FFT-4096 x2)
// Requires ws_size >= ~45 MB.
// ---------------------------------------------------------------------------

#define N_ST   64
#define H_CH   512
#define L_SEQ  4096
#define BATCHN 16
#define STEPSZ 0.001f
#define PI_F   3.14159265358979323846f
#define TWO_PI 6.28318530717958647692f
#define KFS    4104   // stride for Kf rows (>= 4097, 8-aligned)

typedef __attribute__((ext_vector_type(2))) float v2f;
typedef __attribute__((ext_vector_type(8))) float v8f;

// workspace offsets in floats
#define OFF_DRE   ((size_t)0)          // D^T real, [n][l] : 64*4096
#define OFF_DIM   ((size_t)262144)
#define OFF_CRE   ((size_t)524288)     // c[l] : 4096
#define OFF_CIM   ((size_t)528384)
#define OFF_SRE   ((size_t)532480)     // s[l] = k10/(1+k11) : 4096
#define OFF_SIM   ((size_t)536576)
#define OFF_A0RE  ((size_t)540672)     // conj(C)*B, [h][n] : 512*64
#define OFF_A0IM  ((size_t)573440)
#define OFF_A1RE  ((size_t)606208)     // conj(C)*P
#define OFF_A1IM  ((size_t)638976)
#define OFF_ATRE  ((size_t)671744)     // at_roots, [h][l] : 512*4096
#define OFF_ATIM  ((size_t)2768896)
#define OFF_KT    ((size_t)4866048)    // time kernel K, [h][l]
#define OFF_KFRE  ((size_t)6963200)    // Kf, [h][KFS], entries 0..4096 valid
#define OFF_KFIM  ((size_t)9064448)

// -------------------------- small complex helpers ---------------------------
__device__ __forceinline__ float2 cmulf(float2 a, float2 b) {
    return make_float2(a.x * b.x - a.y * b.y, a.x * b.y + a.y * b.x);
}

// ------------------- Phase 0a: per-frequency prep (l loop) ------------------
__global__ __launch_bounds__(256) void s4_prep(
    const float* __restrict__ Lre, const float* __restrict__ Lim,
    const float* __restrict__ Pre, const float* __restrict__ Pim,
    const float* __restrict__ Bre, const float* __restrict__ Bim,
    float* __restrict__ ws)
{
    int l = blockIdx.x * blockDim.x + threadIdx.x;
    if (l >= L_SEQ) return;

    float th = -TWO_PI * (float)l / (float)L_SEQ;
    float osn, ocs;
    __sincosf(th, &osn, &ocs);            // Omega = ocs + i*osn
    float ar = 1.0f - ocs, ai = -osn;     // 1 - Omega
    float br = 1.0f + ocs, bi = osn;      // 1 + Omega
    float den = br * br + bi * bi;
    float gr = (2.0f / STEPSZ) * (ar * br + ai * bi) / den;
    float gi = (2.0f / STEPSZ) * (ai * br - ar * bi) / den;
    float cr = 2.0f * br / den, ci = -2.0f * bi / den;   // c = 2/(1+Omega)

    float k10r = 0.f, k10i = 0.f, k11r = 0.f, k11i = 0.f;
    for (int n = 0; n < N_ST; ++n) {
        float dr = gr - Lre[n], di = gi - Lim[n];
        float dd = dr * dr + di * di;
        float inr = dr / dd, ini = -di / dd;             // 1/(g - lambda)
        ws[OFF_DRE + (size_t)n * L_SEQ + l] = inr;
        ws[OFF_DIM + (size_t)n * L_SEQ + l] = ini;
        // conj(P)*B
        float pbr = Pre[n] * Bre[n] + Pim[n] * Bim[n];
        float pbi = Pre[n] * Bim[n] - Pim[n] * Bre[n];
        k10r += pbr * inr - pbi * ini;
        k10i += pbr * ini + pbi * inr;
        // conj(P)*P = |P|^2 (real)
        float ppr = Pre[n] * Pre[n] + Pim[n] * Pim[n];
        k11r += ppr * inr;
        k11i += ppr * ini;
    }
    float qr = 1.0f + k11r, qi = k11i;
    float qd = qr * qr + qi * qi;
    ws[OFF_SRE + l] = (k10r * qr + k10i * qi) / qd;
    ws[OFF_SIM + l] = (k10i * qr - k10r * qi) / qd;
    ws[OFF_CRE + l] = cr;
    ws[OFF_CIM + l] = ci;
}

// ------------------- Phase 0b: A0 = conj(C)*B, A1 = conj(C)*P ---------------
__global__ __launch_bounds__(256) void s4_a0a1(
    const float* __restrict__ Pre, const float* __restrict__ Pim,
    const float* __restrict__ Bre, const float* __restrict__ Bim,
    const float* __restrict__ Cre, const float* __restrict__ Cim,
    float* __restrict__ ws)
{
    int idx = blockIdx.x * blockDim.x + threadIdx.x;
    if (idx >= H_CH * N_ST) return;
    int n = idx & (N_ST - 1);
    float cre = Cre[idx], cim = Cim[idx];
    ws[OFF_A0RE + idx] = cre * Bre[n] + cim * Bim[n];
    ws[OFF_A0IM + idx] = cre * Bim[n] - cim * Bre[n];
    ws[OFF_A1RE + idx] = cre * Pre[n] + cim * Pim[n];
    ws[OFF_A1IM + idx] = cre * Pim[n] - cim * Pre[n];
}

// ------------- Phase A: Cauchy GEMM + epilogue via fp32 WMMA ----------------
// One wave (32 lanes) computes a 16x16 complex tile of at_roots.
// k00/k01 are complex GEMMs (M=16,N=16,K=64) done as 4 real WMMA chains each.
__global__ __launch_bounds__(256) void s4_cauchy_wmma(float* __restrict__ ws)
{
    const int lane = threadIdx.x & 31;
    const int wave = threadIdx.x >> 5;
    const int tile = blockIdx.x * 8 + wave;   // 8192 tiles total
    const int ht = tile & 31;                 // 32 h-tiles
    const int lt = tile >> 5;                 // 256 l-tiles
    const int half = lane >> 4;               // 0: K lo pair, 1: K hi pair
    const int mloc = lane & 15;

    const float* __restrict__ A0re = ws + OFF_A0RE;
    const float* __restrict__ A0im = ws + OFF_A0IM;
    const float* __restrict__ A1re = ws + OFF_A1RE;
    const float* __restrict__ A1im = ws + OFF_A1IM;
    const float* __restrict__ Dre  = ws + OFF_DRE;
    const float* __restrict__ Dim  = ws + OFF_DIM;

    v8f c00r = {}; v8f c00i = {}; v8f c01r = {}; v8f c01i = {};

    const int hrow = ht * 16 + mloc;          // A-matrix row (M = lane&15)
    const int lcol = lt * 16 + mloc;          // B-matrix col (N = lane&15)

    for (int kk = 0; kk < 16; ++kk) {
        const int k0 = kk * 4 + 2 * half;     // this lane's K pair
        const size_t aidx = (size_t)hrow * N_ST + k0;
        v2f a0r, a0i, a1r, a1i;
        a0r[0] = A0re[aidx]; a0r[1] = A0re[aidx + 1];
        a0i[0] = A0im[aidx]; a0i[1] = A0im[aidx + 1];
        a1r[0] = A1re[aidx]; a1r[1] = A1re[aidx + 1];
        a1i[0] = A1im[aidx]; a1i[1] = A1im[aidx + 1];
        v2f n0i = -a0i, n1i = -a1i;           // fp32 WMMA has no A/B neg bits

        const size_t bidx = (size_t)k0 * L_SEQ + lcol;
        v2f bre, bim;
        bre[0] = Dre[bidx]; bre[1] = Dre[bidx + L_SEQ];
        bim[0] = Dim[bidx]; bim[1] = Dim[bidx + L_SEQ];

        // k00 += (a0r + i a0i)(br + i bi)
        c00r = __builtin_amdgcn_wmma_f32_16x16x4_f32(false, a0r, false, bre, (short)0, c00r, false, false);
        c00r = __builtin_amdgcn_wmma_f32_16x16x4_f32(false, n0i, false, bim, (short)0, c00r, false, false);
        c00i = __builtin_amdgcn_wmma_f32_16x16x4_f32(false, a0r, false, bim, (short)0, c00i, false, false);
        c00i = __builtin_amdgcn_wmma_f32_16x16x4_f32(false, a0i, false, bre, (short)0, c00i, false, false);
        // k01 += (a1r + i a1i)(br + i bi)
        c01r = __builtin_amdgcn_wmma_f32_16x16x4_f32(false, a1r, false, bre, (short)0, c01r, false, false);
        c01r = __builtin_amdgcn_wmma_f32_16x16x4_f32(false, n1i, false, bim, (short)0, c01r, false, false);
        c01i = __builtin_amdgcn_wmma_f32_16x16x4_f32(false, a1r, false, bim, (short)0, c01i, false, false);
        c01i = __builtin_amdgcn_wmma_f32_16x16x4_f32(false, a1i, false, bre, (short)0, c01i, false, false);
    }

    // epilogue: at = c * (k00 - k01 * s)
    const int l = lt * 16 + mloc;
    const float cr = ws[OFF_CRE + l], ci = ws[OFF_CIM + l];
    const float sr = ws[OFF_SRE + l], si = ws[OFF_SIM + l];
    float* __restrict__ atr = ws + OFF_ATRE;
    float* __restrict__ ati = ws + OFF_ATIM;
#pragma unroll
    for (int v = 0; v < 8; ++v) {
        const int m  = v + 8 * half;          // C/D layout: M = v + 8*(lane>=16)
        const int hh = ht * 16 + m;
        float tr = c00r[v] - (c01r[v] * sr - c01i[v] * si);
        float ti = c00i[v] - (c01r[v] * si + c01i[v] * sr);
        atr[(size_t)hh * L_SEQ + l] = cr * tr - ci * ti;
        ati[(size_t)hh * L_SEQ + l] = cr * ti + ci * tr;
    }
}

// ------------------- Stockham radix-2 FFT, length 4096, in LDS --------------
// Input in b0; after 12 stages result lands back in b0 (natural order).
// sign = -1: forward DFT; sign = +1: inverse DFT kernel (no 1/N scaling).
__device__ __forceinline__ void fft4096(float2* b0, float2* b1, int tid, float sign)
{
    float2* src = b0;
    float2* dst = b1;
    int n = 4096;
    int log2s = 0;
    for (int st = 0; st < 12; ++st) {
        const int m = n >> 1;
        const int s = 1 << log2s;
        const float ang = sign * TWO_PI / (float)n;
        for (int i = tid; i < 2048; i += 256) {
            const int q = i & (s - 1);
            const int p = i >> log2s;
            float2 a = src[q + (p << log2s)];
            float2 b = src[q + ((p + m) << log2s)];
            float sn, cs;
            __sincosf(ang * (float)p, &sn, &cs);
            float2 dif = make_float2(a.x - b.x, a.y - b.y);
            dst[q + ((2 * p) << log2s)]     = make_float2(a.x + b.x, a.y + b.y);
            dst[q + ((2 * p + 1) << log2s)] = make_float2(dif.x * cs - dif.y * sn,
                                                          dif.x * sn + dif.y * cs);
        }
        __syncthreads();
        float2* t = src; src = dst; dst = t;
        n >>= 1;
        ++log2s;
    }
}

// Xe/Xo untangle of a packed real-8192 spectrum: X[m] from Z[m&4095], Z[(4096-m)&4095]
__device__ __forceinline__ float2 rfft_untangle(float2 Zm, float2 Zc, float theta)
{
    float xer = 0.5f * (Zm.x + Zc.x);
    float xei = 0.5f * (Zm.y - Zc.y);
    float u = Zm.x - Zc.x, v = Zm.y + Zc.y;
    float xor_ = 0.5f * v, xoi = -0.5f * u;   // (Zm - conj(Zc)) / (2i)
    float sn, cs;
    __sincosf(theta, &sn, &cs);               // theta = -pi*m/4096
    return make_float2(xer + cs * xor_ - sn * xoi,
                       xei + cs * xoi + sn * xor_);
}

// ------------- Phase B: K[h] = Re(ifft(at_roots[h], 4096)) ------------------
__global__ __launch_bounds__(256) void s4_ifft_rows(float* __restrict__ ws)
{
    __shared__ float2 bA[4096];
    __shared__ float2 bB[4096];
    const int h = blockIdx.x;
    const int tid = threadIdx.x;
    const float* __restrict__ ar = ws + OFF_ATRE + (size_t)h * L_SEQ;
    const float* __restrict__ ai = ws + OFF_ATIM + (size_t)h * L_SEQ;
    for (int i = tid; i < 4096; i += 256) bA[i] = make_float2(ar[i], ai[i]);
    __syncthreads();
    fft4096(bA, bB, tid, +1.0f);
    float* __restrict__ kt = ws + OFF_KT + (size_t)h * L_SEQ;
    for (int i = tid; i < 4096; i += 256) kt[i] = bA[i].x * (1.0f / 4096.0f);
}

// ------------- Phase C: Kf[h] = rfft(K[h], 8192) via packed FFT-4096 --------
__global__ __launch_bounds__(256) void s4_rfft_k(float* __restrict__ ws)
{
    __shared__ float2 bA[4096];
    __shared__ float2 bB[4096];
    const int h = blockIdx.x;
    const int tid = threadIdx.x;
    const float* __restrict__ kt = ws + OFF_KT + (size_t)h * L_SEQ;
    for (int t = tid; t < 4096; t += 256)
        bA[t] = (t < 2048) ? make_float2(kt[2 * t], kt[2 * t + 1])
                           : make_float2(0.f, 0.f);
    __syncthreads();
    fft4096(bA, bB, tid, -1.0f);
    float* __restrict__ kfr = ws + OFF_KFRE + (size_t)h * KFS;
    float* __restrict__ kfi = ws + OFF_KFIM + (size_t)h * KFS;
    for (int k = tid; k <= 4096; k += 256) {
        float2 Za = bA[k & 4095];
        float2 Zb = bA[(4096 - k) & 4095];
        float2 X = rfft_untangle(Za, Zb, -PI_F * (float)k / 4096.0f);
        kfr[k] = X.x;
        kfi[k] = X.y;
    }
}

// ------- Phase D: per-row y = irfft(rfft(x,8192) * Kf, 8192)[0:4096] --------
__global__ __launch_bounds__(256) void s4_conv(
    const float* __restrict__ x, const float* __restrict__ ws,
    float* __restrict__ y)
{
    __shared__ float2 bA[4096];
    __shared__ float2 bB[4096];
    const int row = blockIdx.x;               // b*H_CH + h
    const int h = row & (H_CH - 1);
    const int tid = threadIdx.x;

    const float* __restrict__ kfr = ws + OFF_KFRE + (size_t)h * KFS;
    const float* __restrict__ kfi = ws + OFF_KFIM + (size_t)h * KFS;
    __builtin_prefetch(kfr + tid * 16, 0, 0); // global_prefetch_b8 (gfx1250)

    const float* __restrict__ xr = x + (size_t)row * L_SEQ;
    for (int t = tid; t < 4096; t += 256)
        bA[t] = (t < 2048) ? make_float2(xr[2 * t], xr[2 * t + 1])
                           : make_float2(0.f, 0.f);
    __syncthreads();
    fft4096(bA, bB, tid, -1.0f);              // packed spectrum Z in bA

    // untangle -> multiply by Kf -> re-tangle into Z2 (bB)
    for (int k = tid; k <= 2048; k += 256) {
        const int km = 4096 - k;
        float2 Za = bA[k & 4095];
        float2 Zb = bA[km & 4095];
        float2 Xa = rfft_untangle(Za, Zb, -PI_F * (float)k / 4096.0f);
        float2 Xb = rfft_untangle(Zb, Za, -PI_F * (float)km / 4096.0f);
        float2 Ya = cmulf(Xa, make_float2(kfr[k], kfi[k]));
        float2 Yb = cmulf(Xb, make_float2(kfr[km], kfi[km]));

        // Z2[k] = E[k] + i*O[k];  E = (Ya + conj(Yb))/2,
        // O = (Ya - conj(Yb))/2 * e^{+i*pi*k/4096}
        {
            float2 E = make_float2(0.5f * (Ya.x + Yb.x), 0.5f * (Ya.y - Yb.y));
            float2 T = make_float2(0.5f * (Ya.x - Yb.x), 0.5f * (Ya.y + Yb.y));
            float sn, cs;
            __sincosf(PI_F * (float)k / 4096.0f, &sn, &cs);
            float2 O = make_float2(T.x * cs - T.y * sn, T.x * sn + T.y * cs);
            bB[k] = make_float2(E.x - O.y, E.y + O.x);
        }
        if (k != 0 && k != 2048) {            // mirror entry Z2[4096-k]
            float2 E = make_float2(0.5f * (Yb.x + Ya.x), 0.5f * (Yb.y - Ya.y));
            float2 T = make_float2(0.5f * (Yb.x - Ya.x), 0.5f * (Yb.y + Ya.y));
            float sn, cs;
            __sincosf(PI_F * (float)km / 4096.0f, &sn, &cs);
            float2 O = make_float2(T.x * cs - T.y * sn, T.x * sn + T.y * cs);
            bB[km] = make_float2(E.x - O.y, E.y + O.x);
        }
    }
    __syncthreads();
    fft4096(bB, bA, tid, +1.0f);              // inverse; result in bB

    float* __restrict__ yr = y + (size_t)row * L_SEQ;
    for (int t = tid; t < 2048; t += 256) {
        float2 z = bB[t];
        yr[2 * t]     = z.x * (1.0f / 4096.0f);
        yr[2 * t + 1] = z.y * (1.0f / 4096.0f);
    }
}

// ---------------------------------------------------------------------------
extern "C" void kernel_launch(void* const* d_in, const int* in_sizes, int n_in,
                              void* d_out, int out_size, void* d_ws, size_t ws_size,
                              hipStream_t stream)
{
    (void)in_sizes; (void)n_in; (void)out_size; (void)ws_size;
    const float* x   = (const float*)d_in[0];
    const float* Lre = (const float*)d_in[1];
    const float* Lim = (const float*)d_in[2];
    const float* Pre = (const float*)d_in[3];
    const float* Pim = (const float*)d_in[4];
    const float* Bre = (const float*)d_in[5];
    const float* Bim = (const float*)d_in[6];
    const float* Cre = (const float*)d_in[7];
    const float* Cim = (const float*)d_in[8];
    float* y  = (float*)d_out;
    float* ws = (float*)d_ws;

    s4_prep<<<L_SEQ / 256, 256, 0, stream>>>(Lre, Lim, Pre, Pim, Bre, Bim, ws);
    s4_a0a1<<<(H_CH * N_ST) / 256, 256, 0, stream>>>(Pre, Pim, Bre, Bim, Cre, Cim, ws);
    s4_cauchy_wmma<<<1024, 256, 0, stream>>>(ws);          // 8192 tiles / 8 waves
    s4_ifft_rows<<<H_CH, 256, 0, stream>>>(ws);
    s4_rfft_k<<<H_CH, 256, 0, stream>>>(ws);
    s4_conv<<<BATCHN * H_CH, 256, 0, stream>>>(x, ws, y);
}